// DiffNet_72945724555837
// MI455X (gfx1250) — compile-verified
//
#include <hip/hip_runtime.h>
#include <math.h>

typedef __attribute__((ext_vector_type(2))) float v2f;
typedef __attribute__((ext_vector_type(8))) float v8f;

// ---------------------------------------------------------------------------
// Degree: deg[row] += val  (eps added at use sites, matching reference)
// ---------------------------------------------------------------------------
__global__ void diffnet_degree(const int* __restrict__ rows,
                               const float* __restrict__ vals,
                               float* __restrict__ deg, long long nE) {
    long long t = (long long)blockIdx.x * blockDim.x + threadIdx.x;
    if (t < nE) atomicAdd(&deg[rows[t]], vals[t]);
}

// ---------------------------------------------------------------------------
// COO SpMM scatter: out[row] += val * feats[col].  16 threads per edge, each
// thread moves a float4 (coalesced 256B row per half-wave). Accumulators are
// 25.6MB -> resident in the 192MB L2, so these atomics are L2-local.
// ---------------------------------------------------------------------------
__global__ void diffnet_spmm(const int* __restrict__ rows,
                             const int* __restrict__ cols,
                             const float* __restrict__ vals,
                             const float* __restrict__ feats,
                             float* __restrict__ out, long long nE) {
    long long t = (long long)blockIdx.x * blockDim.x + threadIdx.x;
    long long e = t >> 4;
    if (e >= nE) return;
    int j = (int)(t & 15);
    int r = rows[e];
    int c = cols[e];
    float v = vals[e];
    const float4 f = *reinterpret_cast<const float4*>(feats + (size_t)c * 64 + (size_t)j * 4);
    float* o = out + (size_t)r * 64 + (size_t)j * 4;
    atomicAdd(o + 0, v * f.x);
    atomicAdd(o + 1, v * f.y);
    atomicAdd(o + 2, v * f.z);
    atomicAdd(o + 3, v * f.w);
}

// acc = acc / (deg[row]+eps) + h     (GCN layer:  H' = D^-1 A H + H)
__global__ void diffnet_gcn_combine(float* __restrict__ acc,
                                    const float* __restrict__ h,
                                    const float* __restrict__ deg, long long n) {
    long long t = (long long)blockIdx.x * blockDim.x + threadIdx.x;
    if (t < n) {
        int row = (int)(t >> 6);
        acc[t] = acc[t] / (deg[row] + 1e-8f) + h[t];
    }
}

// buf2 = 2*h1 + buf2/(deg_soc+eps) + acc_info/(deg_info+eps)   (final_user)
__global__ void diffnet_final_user(float* __restrict__ buf2,
                                   const float* __restrict__ h1,
                                   const float* __restrict__ accinfo,
                                   const float* __restrict__ dsoc,
                                   const float* __restrict__ dinfo, long long n) {
    long long t = (long long)blockIdx.x * blockDim.x + threadIdx.x;
    if (t < n) {
        int row = (int)(t >> 6);
        buf2[t] = 2.0f * h1[t] + buf2[t] / (dsoc[row] + 1e-8f)
                + accinfo[t] / (dinfo[row] + 1e-8f);
    }
}

// ---------------------------------------------------------------------------
// Scoring via V_WMMA_F32_16X16X4_F32: one wave32 per 16 batch elements.
// C = U(16x64) * V^T(64x16) accumulated over 16 K-chunks of 4; we read the
// diagonal.  A (16x4) layout: lanes 0-15 hold M=lane, VGPR0/1 = K0/K1;
// lanes 16-31 = K2/K3.  B (4x16) is the mirrored layout, so both operands
// use the identical per-lane gather: row = lane&15, cols = 4*kk + 2*(lane>>4).
// Diagonal m: m<8 -> c[m] on lane m; m>=8 -> c[m-8] on lane m+16.
// ---------------------------------------------------------------------------
__global__ __launch_bounds__(32) void diffnet_score_wmma(
        const float* __restrict__ fu, const float* __restrict__ item,
        const int* __restrict__ uid, const int* __restrict__ iid,
        float* __restrict__ out, int batch) {
    int lane = (int)threadIdx.x;
    int r    = lane & 15;
    int half = lane >> 4;
    int b    = (int)blockIdx.x * 16 + r;
    int bc   = b < batch ? b : (batch - 1);   // clamp: keep EXEC all-ones for WMMA

    const float* urow = fu   + (size_t)uid[bc] * 64;
    const float* vrow = item + (size_t)iid[bc] * 64;

    v8f c = {};
#pragma unroll
    for (int kk = 0; kk < 16; ++kk) {
        int col = kk * 4 + half * 2;
        v2f a, bm;
        a.x  = urow[col];
        a.y  = urow[col + 1];
        bm.x = 2.0f * vrow[col];        // final_item = 2 * item_emb
        bm.y = 2.0f * vrow[col + 1];
        c = __builtin_amdgcn_wmma_f32_16x16x4_f32(
                /*neg_a=*/false, a, /*neg_b=*/false, bm,
                /*c_mod=*/(short)0, c, /*reuse_a=*/false, /*reuse_b=*/false);
    }

    // Extract this lane's diagonal component.
    int comp = lane & 7;
    float d = c[0];
    d = (comp == 1) ? c[1] : d;
    d = (comp == 2) ? c[2] : d;
    d = (comp == 3) ? c[3] : d;
    d = (comp == 4) ? c[4] : d;
    d = (comp == 5) ? c[5] : d;
    d = (comp == 6) ? c[6] : d;
    d = (comp == 7) ? c[7] : d;

    bool active = (lane < 8) || (lane >= 24);
    int  m  = (lane < 8) ? lane : (lane - 16);
    int  ob = (int)blockIdx.x * 16 + m;
    if (active && ob < batch)
        out[ob] = 1.0f / (1.0f + __expf(-d));
}

// ---------------------------------------------------------------------------
extern "C" void kernel_launch(void* const* d_in, const int* in_sizes, int n_in,
                              void* d_out, int out_size, void* d_ws, size_t ws_size,
                              hipStream_t stream) {
    const float* user_emb = (const float*)d_in[0];
    const float* item_emb = (const float*)d_in[1];
    const int*   s_rows   = (const int*)d_in[2];
    const int*   s_cols   = (const int*)d_in[3];
    const float* s_vals   = (const float*)d_in[4];
    const int*   i_rows   = (const int*)d_in[5];
    const int*   i_cols   = (const int*)d_in[6];
    const float* i_vals   = (const float*)d_in[7];
    const int*   uid      = (const int*)d_in[8];
    const int*   iid      = (const int*)d_in[9];
    float*       out      = (float*)d_out;

    const long long n_user = in_sizes[0] / 64;
    const long long e_soc  = in_sizes[2];
    const long long e_info = in_sizes[5];
    const int       batch  = in_sizes[8];
    const long long nfeat  = n_user * 64;

    // Workspace layout (floats): deg_info | deg_soc | acc_info | buf1 | buf2
    float* deg_info = (float*)d_ws;
    float* deg_soc  = deg_info + n_user;
    float* acc_info = deg_soc  + n_user;
    float* buf1     = acc_info + nfeat;      // spmm(social, user_emb) -> h1
    float* buf2     = buf1     + nfeat;      // spmm(social, h1) -> final_user

    size_t zero_bytes = (size_t)(2 * n_user + 3 * nfeat) * sizeof(float);
    hipMemsetAsync(d_ws, 0, zero_bytes, stream);

    const int T = 256;
    dim3 blk(T);

    // Degrees
    diffnet_degree<<<dim3((unsigned)((e_info + T - 1) / T)), blk, 0, stream>>>(
        i_rows, i_vals, deg_info, e_info);
    diffnet_degree<<<dim3((unsigned)((e_soc + T - 1) / T)), blk, 0, stream>>>(
        s_rows, s_vals, deg_soc, e_soc);

    // user_from_items numerator: acc_info = info_adj @ item_emb
    diffnet_spmm<<<dim3((unsigned)((e_info * 16 + T - 1) / T)), blk, 0, stream>>>(
        i_rows, i_cols, i_vals, item_emb, acc_info, e_info);

    // h1 = D^-1 A user_emb + user_emb
    diffnet_spmm<<<dim3((unsigned)((e_soc * 16 + T - 1) / T)), blk, 0, stream>>>(
        s_rows, s_cols, s_vals, user_emb, buf1, e_soc);
    diffnet_gcn_combine<<<dim3((unsigned)((nfeat + T - 1) / T)), blk, 0, stream>>>(
        buf1, user_emb, deg_soc, nfeat);

    // h2 numerator: buf2 = A h1 ; then final_user = 2*h1 + buf2/deg + acc/deg_info
    diffnet_spmm<<<dim3((unsigned)((e_soc * 16 + T - 1) / T)), blk, 0, stream>>>(
        s_rows, s_cols, s_vals, buf1, buf2, e_soc);
    diffnet_final_user<<<dim3((unsigned)((nfeat + T - 1) / T)), blk, 0, stream>>>(
        buf2, buf1, acc_info, deg_soc, deg_info, nfeat);

    // Scores: sigmoid(diag(U V^T)) via f32 WMMA
    diffnet_score_wmma<<<dim3((unsigned)((batch + 15) / 16)), dim3(32), 0, stream>>>(
        buf2, item_emb, uid, iid, out, batch);
}